// DenseGrid_16381005267401
// MI455X (gfx1250) — compile-verified
//
#include <hip/hip_runtime.h>
#include <cstdint>

// ---------------------------------------------------------------------------
// Multi-resolution dense grid bilinear sampler for MI455X (gfx1250).
// Memory-bound (268 MB output stream) -> optimize data movement only:
//   - codebooks re-laid-out to AoS float4 (one global_load_b128 per corner)
//   - levels 0..2 (62.5 KB) staged into LDS via global_load_async_to_lds_b128
//     (ASYNCcnt + s_wait_asynccnt), gathered with ds_load_b128
//   - non-temporal b128 output stores: write-once stream must not evict the
//     28.5 MB codebook working set from the 192 MB L2
//   - global_prefetch_b8 issued early for the 20 MB level-7 random gathers
// ---------------------------------------------------------------------------

namespace {
constexpr int kNumLevels  = 8;
constexpr int kRes[kNumLevels] = {16, 29, 53, 98, 181, 332, 610, 1120};
// exclusive prefix sum of res*res
constexpr int kOff[kNumLevels] = {0, 256, 1097, 3906, 13510, 46271, 156495, 528595};
constexpr int kTotalCells = 1782995;       // sum of res*res over all levels
constexpr int kLdsCells   = 3906;          // levels 0..2 -> 62,496 B of LDS
constexpr int kSamples    = 4 * 524288;    // BATCH * NUM_SAMPLES
}

struct CbPtrs { const float* p[kNumLevels]; };

// vector types
typedef float v4f __attribute__((ext_vector_type(4)));
typedef float v2f __attribute__((ext_vector_type(2)));
// 16-byte int vector type matching the async-LDS builtin's pointee type.
typedef int v4i_t __attribute__((vector_size(16)));
typedef __attribute__((address_space(1))) v4i_t* v4i_global_ptr;
typedef __attribute__((address_space(3))) v4i_t* v4i_lds_ptr;

__device__ __forceinline__ v4f bilerp4(v4f f00, v4f f01, v4f f10, v4f f11,
                                       float wx, float wy) {
    const float w00 = (1.0f - wx) * (1.0f - wy);
    const float w01 = wx * (1.0f - wy);
    const float w10 = (1.0f - wx) * wy;
    const float w11 = wx * wy;
    return f00 * w00 + f01 * w01 + f10 * w10 + f11 * w11;
}

// ---------------------------------------------------------------------------
// Main kernel (workspace path): one thread per sample, 8 levels unrolled,
// 128 B contiguous NT output per thread.  Defined FIRST so the disasm
// snippet shows this kernel.
// ---------------------------------------------------------------------------
__global__ __launch_bounds__(1024)
void densegrid_ws_kernel(const float* __restrict__ coords,
                         const float4* __restrict__ ws4,
                         float* __restrict__ out,
                         int nlev)
{
    __shared__ float4 sh[kLdsCells];   // transposed levels 0..2

    // ---- Stage levels 0..2 into LDS via the async copy engine ------------
#if __has_builtin(__builtin_amdgcn_global_load_async_to_lds_b128)
#pragma unroll
    for (int k = 0; k < 4; ++k) {
        const int i = (int)threadIdx.x + k * 1024;
        if (i < kLdsCells) {
            __builtin_amdgcn_global_load_async_to_lds_b128(
                (v4i_global_ptr)(uintptr_t)(ws4 + i),
                (v4i_lds_ptr)(sh + i),
                /*offset=*/0, /*cpol=*/0);
        }
    }
#if __has_builtin(__builtin_amdgcn_s_wait_asynccnt)
    __builtin_amdgcn_s_wait_asynccnt(0);
#else
    asm volatile("s_wait_asynccnt 0" ::: "memory");
#endif
#else
    for (int i = threadIdx.x; i < kLdsCells; i += 1024)
        sh[i] = ws4[i];
#endif
    __syncthreads();

    const int tid = blockIdx.x * blockDim.x + threadIdx.x;
    if (tid >= kSamples) return;

    const v2f c = __builtin_nontemporal_load((const v2f*)coords + tid);
    const float bx = (c.x + 1.0f) * 0.5f;   // hoisted: px = bx*(R-1)
    const float by = (c.y + 1.0f) * 0.5f;

    // Early prefetch of the largest level's corner lines (20 MB table ->
    // random gather; overlap its latency with the small-level math).
    if (nlev > 7) {
        const int   R  = kRes[7];
        const float px = bx * (float)(R - 1);
        const float py = by * (float)(R - 1);
        int x0 = (int)floorf(px), y0 = (int)floorf(py);
        x0 = min(max(x0, 0), R - 1);
        y0 = min(max(y0, 0), R - 1);
        const int x1 = min(x0 + 1, R - 1);
        const int y1 = min(y0 + 1, R - 1);
        const char* b = (const char*)(ws4 + kOff[7]);
        __builtin_prefetch(b + (size_t)(y0 * R + x0) * 16, 0, 0);
        __builtin_prefetch(b + (size_t)(y0 * R + x1) * 16, 0, 0);
        __builtin_prefetch(b + (size_t)(y1 * R + x0) * 16, 0, 0);
        __builtin_prefetch(b + (size_t)(y1 * R + x1) * 16, 0, 0);
    }

    float* __restrict__ o = out + (size_t)tid * (size_t)(nlev * 4);

#pragma unroll
    for (int l = 0; l < kNumLevels; ++l) {
        if (l < nlev) {
            const int   R  = kRes[l];
            const float px = bx * (float)(R - 1);
            const float py = by * (float)(R - 1);
            const float xf = floorf(px);
            const float yf = floorf(py);
            const float wx = px - xf;
            const float wy = py - yf;
            const int x0 = min(max((int)xf, 0), R - 1);
            const int x1 = min(max((int)xf + 1, 0), R - 1);
            const int y0 = min(max((int)yf, 0), R - 1);
            const int y1 = min(max((int)yf + 1, 0), R - 1);

            const v4f* __restrict__ b = (l < 3)
                ? (const v4f*)sh  + kOff[l]     // ds_load_b128 gathers
                : (const v4f*)ws4 + kOff[l];    // global_load_b128 gathers
            const v4f f00 = b[y0 * R + x0];
            const v4f f01 = b[y0 * R + x1];
            const v4f f10 = b[y1 * R + x0];
            const v4f f11 = b[y1 * R + x1];

            const v4f r = bilerp4(f00, f01, f10, f11, wx, wy);
            // NT store: 8 per thread fill one full 128B line; never re-read.
            __builtin_nontemporal_store(r, (v4f*)(o + 4 * l));
        }
    }
}

// ---------------------------------------------------------------------------
// Fallback (workspace too small): gather straight from the SoA planes.
// ---------------------------------------------------------------------------
__global__ __launch_bounds__(1024)
void densegrid_nows_kernel(const float* __restrict__ coords,
                           CbPtrs cbs,
                           float* __restrict__ out,
                           int nlev)
{
    __shared__ float4 sh[kLdsCells];   // transposed levels 0..2

    for (int i = threadIdx.x; i < kLdsCells; i += 1024) {
        const int lev = (i >= kOff[2]) ? 2 : (i >= kOff[1] ? 1 : 0);
        const int ci  = i - kOff[lev];
        const int r2  = kRes[lev] * kRes[lev];
        const float* __restrict__ cb = cbs.p[lev];
        sh[i] = make_float4(cb[ci], cb[r2 + ci], cb[2 * r2 + ci], cb[3 * r2 + ci]);
    }
    __syncthreads();

    const int tid = blockIdx.x * blockDim.x + threadIdx.x;
    if (tid >= kSamples) return;

    const v2f c = __builtin_nontemporal_load((const v2f*)coords + tid);
    const float bx = (c.x + 1.0f) * 0.5f;
    const float by = (c.y + 1.0f) * 0.5f;

    float* __restrict__ o = out + (size_t)tid * (size_t)(nlev * 4);

#pragma unroll
    for (int l = 0; l < kNumLevels; ++l) {
        if (l < nlev) {
            const int   R  = kRes[l];
            const float px = bx * (float)(R - 1);
            const float py = by * (float)(R - 1);
            const float xf = floorf(px);
            const float yf = floorf(py);
            const float wx = px - xf;
            const float wy = py - yf;
            const int x0 = min(max((int)xf, 0), R - 1);
            const int x1 = min(max((int)xf + 1, 0), R - 1);
            const int y0 = min(max((int)yf, 0), R - 1);
            const int y1 = min(max((int)yf + 1, 0), R - 1);

            v4f f00, f01, f10, f11;
            if (l < 3) {
                const v4f* __restrict__ b = (const v4f*)sh + kOff[l];
                f00 = b[y0 * R + x0];
                f01 = b[y0 * R + x1];
                f10 = b[y1 * R + x0];
                f11 = b[y1 * R + x1];
            } else {
                const float* __restrict__ cb = cbs.p[l];
                const int r2 = R * R;
                const int i00 = y0 * R + x0, i01 = y0 * R + x1;
                const int i10 = y1 * R + x0, i11 = y1 * R + x1;
                f00 = (v4f){cb[i00], cb[r2 + i00], cb[2 * r2 + i00], cb[3 * r2 + i00]};
                f01 = (v4f){cb[i01], cb[r2 + i01], cb[2 * r2 + i01], cb[3 * r2 + i01]};
                f10 = (v4f){cb[i10], cb[r2 + i10], cb[2 * r2 + i10], cb[3 * r2 + i10]};
                f11 = (v4f){cb[i11], cb[r2 + i11], cb[2 * r2 + i11], cb[3 * r2 + i11]};
            }

            const v4f r = bilerp4(f00, f01, f10, f11, wx, wy);
            __builtin_nontemporal_store(r, (v4f*)(o + 4 * l));
        }
    }
}

// ---------------------------------------------------------------------------
// Pass 1: transpose all codebooks [F=4, R, R] -> [R*R] x float4 into d_ws.
// Branchless level resolution (immediates + cndmask, no memory tables).
// ---------------------------------------------------------------------------
__global__ void transpose_cbs_kernel(CbPtrs cbs, float4* __restrict__ ws4) {
    const int t = blockIdx.x * blockDim.x + threadIdx.x;
    if (t >= kTotalCells) return;
    int off = 0, r2 = kRes[0] * kRes[0];
    const float* cb = cbs.p[0];
#pragma unroll
    for (int l = 1; l < kNumLevels; ++l) {
        if (t >= kOff[l]) {
            off = kOff[l];
            r2  = kRes[l] * kRes[l];
            cb  = cbs.p[l];
        }
    }
    const int c = t - off;
    ws4[t] = make_float4(cb[c], cb[r2 + c], cb[2 * r2 + c], cb[3 * r2 + c]);
}

// ---------------------------------------------------------------------------
extern "C" void kernel_launch(void* const* d_in, const int* in_sizes, int n_in,
                              void* d_out, int out_size, void* d_ws, size_t ws_size,
                              hipStream_t stream) {
    const float* coords = (const float*)d_in[0];
    // d_in[1] is lod_idx; number of active levels is recovered from out_size
    // (deterministic, no device->host read needed).
    CbPtrs cbs;
    for (int i = 0; i < kNumLevels; ++i)
        cbs.p[i] = (const float*)d_in[2 + i];

    float* out = (float*)d_out;
    int nlev = out_size / (kSamples * 4);
    if (nlev < 1) nlev = 1;
    if (nlev > kNumLevels) nlev = kNumLevels;

    const size_t ws_needed = (size_t)kTotalCells * sizeof(float4);
    const int grid_main = (kSamples + 1023) / 1024;

    if (ws_size >= ws_needed) {
        float4* ws4 = (float4*)d_ws;
        transpose_cbs_kernel<<<(kTotalCells + 255) / 256, 256, 0, stream>>>(cbs, ws4);
        densegrid_ws_kernel<<<grid_main, 1024, 0, stream>>>(
            coords, (const float4*)ws4, out, nlev);
    } else {
        densegrid_nows_kernel<<<grid_main, 1024, 0, stream>>>(
            coords, cbs, out, nlev);
    }
}